// TTCircuit_69939247448064
// MI455X (gfx1250) — compile-verified
//
#include <hip/hip_runtime.h>

typedef __attribute__((ext_vector_type(2))) float v2f;
typedef __attribute__((ext_vector_type(4))) float v4f;
typedef __attribute__((ext_vector_type(8))) float v8f;

#define NQ 8
#define R  64
#define DD 16
#define RLB 16
#define ROB 4
#define ENV_N (R*RLB*ROB*RLB*R)   // 4,194,304 floats (16.8 MB)

// ---- LDS layout (float offsets); padded strides kill bank conflicts on A-gathers ----
#define OFF_S    0                    // 16384  S panel   [a][(i*16+Ac)]
#define OFF_O    (OFF_S   + 16384)    // 4096   O panel   [(o*16+j)][(k*4+Oc)]
#define OFF_UD   (OFF_O   + 4096)     // 16384  Ud panel  [(l*16+i)][(j*4+Ll)]
#define OFF_ENV  (OFF_UD  + 16384)    // 4096   env slice [a][(l*4+o)]
#define OFF_T1   (OFF_ENV + 4096)     // 16384  T1 [(l*4+o)][(i*16+Ac)]
#define OFF_T2   (OFF_T1  + 16384)    // 4352   T2 [(o*16+Ac)][pad68: (j*4+Ll)]
#define OFF_T3   (OFF_T2  + 4352)     // 4352   T3 [(Ac*4+Ll)][pad68: (k*4+Oc)]
#define OFF_U4   (OFF_T3  + 4352)     // 1152   U4 [(k*4+Ml)][pad18: s]
#define OFF_SB   (OFF_U4  + 1152)     // 256    Sb [s][Bc]
#define OFF_G    (OFF_SB  + 256)      // 1024   G  [k][(Ml*16+Bc)]
#define SMEM_FLOATS (OFF_G + 1024)    // 68480 floats = 267.5 KB < 320 KB/WGP

__device__ __forceinline__ v8f wmma4(v2f a, v2f b, v8f c) {
  // V_WMMA_F32_16X16X4_F32 : D = A(16x4) x B(4x16) + C(16x16), fp32 native
  return __builtin_amdgcn_wmma_f32_16x16x4_f32(false, a, false, b, (short)0, c,
                                               false, false);
}

// env layout (transposed for contiguous slice loads): envT[m][b][a][(l*4+o)]
__global__ __launch_bounds__(256, 1)
void tt_site_kernel(const float* __restrict__ envIn,
                    float* __restrict__ envOut,
                    const float* __restrict__ S,   // [64][16][64]  (r,d,r)
                    const float* __restrict__ U,   // [16][16][16][16] (rl,d,d,rl)
                    const float* __restrict__ Oq)  // [4][16][16][4]   (ro,d,d,ro)
{
  extern __shared__ float sm[];
  const int tid  = threadIdx.x;
  const int lane = tid & 31;
  const int w    = tid >> 5;            // wave id 0..7
  const int lrow = lane & 15;           // row (A-frag) / col (B-frag) within tile
  const int koff = (lane >> 4) << 1;    // K sub-offset: lanes 0-15 -> {0,1}, 16-31 -> {2,3}
  const int hi8  = (lane >> 4) << 3;    // C/D row offset for upper half-wave

  const int bidx = blockIdx.x;          // 256 WGs: (At, Bt, Mt, Lt) each in [0,4)
  const int At = bidx & 3, Bt = (bidx >> 2) & 3, Mt = (bidx >> 4) & 3, Lt = (bidx >> 6) & 3;

  // ---- one-time resident panels ----
  for (int j = 0; j < 64; ++j) {
    int e = tid + j * 256;              // 0..16383
    { int a = e >> 8, i = (e >> 4) & 15, Ac = e & 15;
      sm[OFF_S + e] = S[(a * 16 + i) * 64 + At * 16 + Ac]; }
    if (e < 4096) sm[OFF_O + e] = Oq[e];   // natural layout matches
    { int li = e >> 6, jl = e & 63;
      int l = li >> 4, i = li & 15, jj = jl >> 2, Ll = jl & 3;
      // Ud[(l,i),(j,L)] = U[l][j][i][L]  (real inputs: conj == identity)
      sm[OFF_UD + e] = U[((l * 16 + jj) * 16 + i) * 16 + Lt * 4 + Ll]; }
  }

  v8f acc[8] = {{}, {}, {}, {}, {}, {}, {}, {}};  // persistent env' acc (8 C-tiles/wave)

  for (int m = 0; m < 16; ++m) {
    // U4[(k*4+Ml)][s] = U[m][k][s][Mt*4+Ml]   (padded stride 18)
    #pragma unroll
    for (int j = 0; j < 4; ++j) {
      int e = tid + j * 256;            // 0..1023
      int row = e >> 4, s = e & 15, k = row >> 2, Ml = row & 3;
      sm[OFF_U4 + row * 18 + s] = U[((m * 16 + k) * 16 + s) * 16 + Mt * 4 + Ml];
    }

    for (int b = 0; b < 64; ++b) {
      {
        int s = tid >> 4, Bc = tid & 15;         // 256 threads -> exactly Sb
        sm[OFF_SB + tid] = S[(b * 16 + s) * 64 + Bt * 16 + Bc];
      }
      {
        const float* src = envIn + (size_t)(m * 64 + b) * 4096;
        #pragma unroll
        for (int j = 0; j < 4; ++j)
          ((v4f*)(sm + OFF_ENV))[tid + j * 256] = ((const v4f*)src)[tid + j * 256];
        if ((m * 64 + b) < 1023)        // prefetch next env slice into caches
          __builtin_prefetch(envIn + (size_t)(m * 64 + b + 1) * 4096 + lane * 16, 0, 1);
      }
      __syncthreads();

      // ---- G1: T1[(l,o)][(i,Ac)] = sum_a env[a][(l,o)] * S[a][(i,Ac)]  (K=64) ----
      // wave owns 1 rowT x 8 colT tiles: A-frag reused by 8 WMMAs per K-step
      {
        int rowT = w >> 1;              // (l,o) row-tile
        int colB = (w & 1) * 8;         // first of 8 (i,Ac) col-tiles
        v8f c[8] = {{}, {}, {}, {}, {}, {}, {}, {}};
        #pragma unroll
        for (int ks = 0; ks < 16; ++ks) {
          int k0 = ks * 4 + koff;
          v2f a;
          a.x = sm[OFF_ENV + (k0    ) * 64 + rowT * 16 + lrow];
          a.y = sm[OFF_ENV + (k0 + 1) * 64 + rowT * 16 + lrow];
          #pragma unroll
          for (int ti = 0; ti < 8; ++ti) {
            v2f bb;
            bb.x = sm[OFF_S + (k0    ) * 256 + (colB + ti) * 16 + lrow];
            bb.y = sm[OFF_S + (k0 + 1) * 256 + (colB + ti) * 16 + lrow];
            c[ti] = wmma4(a, bb, c[ti]);
          }
        }
        #pragma unroll
        for (int ti = 0; ti < 8; ++ti)
          #pragma unroll
          for (int v = 0; v < 8; ++v)
            sm[OFF_T1 + (rowT * 16 + v + hi8) * 256 + (colB + ti) * 16 + lrow] = c[ti][v];
      }
      __syncthreads();

      // ---- G2: T2[(o,Ac)][(j,Ll)] = sum_{l,i} T1[(l,o)][(i,Ac)] * Ud[(l,i)][(j,Ll)]
      //      (K=256; wave owns 1 rowT x 2 colT: shared A-gather, dual K-chains) ----
      {
        int rowT = w >> 1;              // o
        int colB = (w & 1) * 2;         // first of 2 (j,Ll) col-tiles
        v8f c0 = {}, c1 = {};
        #pragma unroll
        for (int ks = 0; ks < 64; ++ks) {
          int k0 = ks * 4 + koff;
          int k1 = k0 + 1;
          v2f a, b0, b1;
          a.x = sm[OFF_T1 + ((k0 >> 4) * 4 + rowT) * 256 + (k0 & 15) * 16 + lrow];
          a.y = sm[OFF_T1 + ((k1 >> 4) * 4 + rowT) * 256 + (k1 & 15) * 16 + lrow];
          b0.x = sm[OFF_UD + k0 * 64 + (colB    ) * 16 + lrow];
          b0.y = sm[OFF_UD + k1 * 64 + (colB    ) * 16 + lrow];
          b1.x = sm[OFF_UD + k0 * 64 + (colB + 1) * 16 + lrow];
          b1.y = sm[OFF_UD + k1 * 64 + (colB + 1) * 16 + lrow];
          c0 = wmma4(a, b0, c0);
          c1 = wmma4(a, b1, c1);
        }
        #pragma unroll
        for (int v = 0; v < 8; ++v) {
          sm[OFF_T2 + (rowT * 16 + v + hi8) * 68 + (colB    ) * 16 + lrow] = c0[v];
          sm[OFF_T2 + (rowT * 16 + v + hi8) * 68 + (colB + 1) * 16 + lrow] = c1[v];
        }
      }
      __syncthreads();

      // ---- G3: T3[(Ac,Ll)][(k,Oc)] = sum_{o,j} T2[(o,Ac)][(j,Ll)] * O[(o,j)][(k,Oc)]
      //      (K=64; wave owns 1 rowT x 2 colT) + G-build on waves 0-3 ----
      {
        int rowT = w >> 1;
        int colB = (w & 1) * 2;
        int row = rowT * 16 + lrow;     // (Ac*4+Ll)
        v8f c0 = {}, c1 = {};
        #pragma unroll
        for (int ks = 0; ks < 16; ++ks) {
          int k0 = ks * 4 + koff;
          int k1 = k0 + 1;
          v2f a, b0, b1;
          a.x = sm[OFF_T2 + ((k0 >> 4) * 16 + (row >> 2)) * 68 + (k0 & 15) * 4 + (row & 3)];
          a.y = sm[OFF_T2 + ((k1 >> 4) * 16 + (row >> 2)) * 68 + (k1 & 15) * 4 + (row & 3)];
          b0.x = sm[OFF_O + k0 * 64 + (colB    ) * 16 + lrow];
          b0.y = sm[OFF_O + k1 * 64 + (colB    ) * 16 + lrow];
          b1.x = sm[OFF_O + k0 * 64 + (colB + 1) * 16 + lrow];
          b1.y = sm[OFF_O + k1 * 64 + (colB + 1) * 16 + lrow];
          c0 = wmma4(a, b0, c0);
          c1 = wmma4(a, b1, c1);
        }
        #pragma unroll
        for (int v = 0; v < 8; ++v) {
          sm[OFF_T3 + (rowT * 16 + v + hi8) * 68 + (colB    ) * 16 + lrow] = c0[v];
          sm[OFF_T3 + (rowT * 16 + v + hi8) * 68 + (colB + 1) * 16 + lrow] = c1[v];
        }

        // G[k][(Ml,Bc)] = sum_s U4[(k,Ml)][s] * Sb[s][Bc]  (K=16, 4 tiles on waves 0-3)
        if (w < 4) {
          int rT = w;                   // row-tile over (k*4+Ml)
          v8f cg = {};
          #pragma unroll
          for (int ks = 0; ks < 4; ++ks) {
            int k0 = ks * 4 + koff;
            v2f a, bb;
            a.x  = sm[OFF_U4 + (rT * 16 + lrow) * 18 + k0];
            a.y  = sm[OFF_U4 + (rT * 16 + lrow) * 18 + k0 + 1];
            bb.x = sm[OFF_SB + k0 * 16 + lrow];
            bb.y = sm[OFF_SB + (k0 + 1) * 16 + lrow];
            cg = wmma4(a, bb, cg);
          }
          #pragma unroll
          for (int v = 0; v < 8; ++v) {
            int grow = rT * 16 + v + hi8;    // = k*4+Ml
            sm[OFF_G + (grow >> 2) * 64 + (grow & 3) * 16 + lrow] = cg[v];
          }
        }
      }
      __syncthreads();

      // ---- Final: acc[(Ac,Ll,Oc)][(Ml,Bc)] += sum_k T3 * G  (K=16, 8 tiles/wave) ----
      // tiles: rowT = w*2 + h (h=0,1), colT = 0..3; B-frags shared across h
      {
        #pragma unroll
        for (int ks = 0; ks < 4; ++ks) {
          int k0 = ks * 4 + koff;
          v2f bb[4];
          #pragma unroll
          for (int ci = 0; ci < 4; ++ci) {
            bb[ci].x = sm[OFF_G + k0 * 64 + ci * 16 + lrow];
            bb[ci].y = sm[OFF_G + (k0 + 1) * 64 + ci * 16 + lrow];
          }
          #pragma unroll
          for (int h = 0; h < 2; ++h) {
            int rowT = w * 2 + h;
            int row = rowT * 16 + lrow;   // (Ac*16 + Ll*4 + Oc)
            int arow = ((row >> 4) * 4 + ((row >> 2) & 3)) * 68 + (row & 3);
            v2f a;
            a.x = sm[OFF_T3 + arow + k0 * 4];
            a.y = sm[OFF_T3 + arow + (k0 + 1) * 4];
            #pragma unroll
            for (int ci = 0; ci < 4; ++ci)
              acc[h * 4 + ci] = wmma4(a, bb[ci], acc[h * 4 + ci]);
          }
        }
      }
      __syncthreads();
    } // b
  } // m

  // ---- store env' slice directly in next-site transposed layout [M][B][A][(L*4+O)] ----
  #pragma unroll
  for (int h = 0; h < 2; ++h) {
    int rowT = w * 2 + h;
    #pragma unroll
    for (int ci = 0; ci < 4; ++ci) {
      #pragma unroll
      for (int v = 0; v < 8; ++v) {
        int row = rowT * 16 + v + hi8;
        int col = ci * 16 + lrow;
        int Ac = row >> 4, Ll = (row >> 2) & 3, Oc = row & 3;
        int Ml = col >> 4, Bc = col & 15;
        int Ag = At * 16 + Ac, Lg = Lt * 4 + Ll, Mg = Mt * 4 + Ml, Bg = Bt * 16 + Bc;
        envOut[((size_t)(Mg * 64 + Bg) * 64 + Ag) * 64 + (Lg * 4 + Oc)] = acc[h * 4 + ci][v];
      }
    }
  }
}

__global__ void init_env_kernel(float* envT) {
  size_t i = (size_t)blockIdx.x * 1024 + threadIdx.x;
  if (i < (size_t)ENV_N) envT[i] = (i == 0) ? 1.0f : 0.0f;  // e0 boundary closure
}

__global__ void extract_kernel(const float* envT, float* out) {
  if (threadIdx.x == 0 && blockIdx.x == 0) out[0] = envT[0];  // [0,0,0,0,0]
}

extern "C" void kernel_launch(void* const* d_in, const int* in_sizes, int n_in,
                              void* d_out, int out_size, void* d_ws, size_t ws_size,
                              hipStream_t stream) {
  const float* state = (const float*)d_in[0];  // [8][64][16][64]
  const float* layer = (const float*)d_in[1];  // [1][8][16][16][16][16]
  const float* oper  = (const float*)d_in[2];  // [8][4][16][16][4]

  float* envA = (float*)d_ws;                  // ping/pong env buffers (2 x 16.8 MB)
  float* envB = envA + ENV_N;

  init_env_kernel<<<ENV_N / 1024, 1024, 0, stream>>>(envA);

  const size_t smem = SMEM_FLOATS * sizeof(float);
  for (int q = 0; q < NQ; ++q) {
    const float* Sq = state + (size_t)q * R * DD * R;
    const float* Uq = layer + (size_t)q * RLB * DD * DD * RLB;  // layer[0][q]
    const float* Oc = oper  + (size_t)q * ROB * DD * DD * ROB;
    float* ein  = (q & 1) ? envB : envA;
    float* eout = (q & 1) ? envA : envB;
    tt_site_kernel<<<256, 256, smem, stream>>>(ein, eout, Sq, Uq, Oc);
  }
  // after q=7 (odd), result is in envA
  extract_kernel<<<1, 1, 0, stream>>>(envA, (float*)d_out);
}